// LCN_56229711839459
// MI455X (gfx1250) — compile-verified
//
#include <hip/hip_runtime.h>
#include <hip/hip_bf16.h>

typedef __attribute__((ext_vector_type(2))) float v2f;
typedef __attribute__((ext_vector_type(8))) float v8f;

// ---------------------------------------------------------------------------
// 1) Transpose x (256 x F, batch-major) -> xT (F x 256, feature-major) so that
//    all subsequent gathers are coalesced across lanes (lane = batch).
// ---------------------------------------------------------------------------
__global__ void lcn_transpose(const float* __restrict__ x,
                              float* __restrict__ xT, int F) {
  __shared__ float tile[32][33];
  const int f0 = blockIdx.x * 32;
  const int b0 = blockIdx.y * 32;
  #pragma unroll
  for (int i = threadIdx.y; i < 32; i += 8)
    tile[i][threadIdx.x] = x[(size_t)(b0 + i) * F + (f0 + threadIdx.x)];
  __syncthreads();
  #pragma unroll
  for (int i = threadIdx.y; i < 32; i += 8)
    xT[(size_t)(f0 + i) * 256 + (b0 + threadIdx.x)] = tile[threadIdx.x][i];
}

// ---------------------------------------------------------------------------
// 2) One LCN layer in feature-major layout.
//    Block = one output feature d (knn/w/bias are wave-uniform -> SGPR loads);
//    thread = batch. Gather actIn[idx*256 + b] is a fully coalesced 128B load
//    per wave. out[b,d] = relu(sum_k actIn[knn[d,k],b] * w[d,k] + bias[d]).
// ---------------------------------------------------------------------------
__global__ void lcn_layer(const float* __restrict__ actIn,  // prev x 256
                          const float* __restrict__ w,      // d x 16
                          const float* __restrict__ bias,   // d
                          const int*   __restrict__ knn,    // d x 16
                          float* __restrict__ actOut) {     // dim x 256
  const int d = blockIdx.x;
  const int b = threadIdx.x;  // 0..255
  float acc = 0.0f;
  #pragma unroll
  for (int k = 0; k < 16; ++k) {
    const int   idx = knn[d * 16 + k];   // uniform -> SGPR
    const float wv  = w  [d * 16 + k];   // uniform -> SGPR
    acc = fmaf(actIn[(size_t)idx * 256 + b], wv, acc);
  }
  const float r = acc + bias[d];
  actOut[(size_t)d * 256 + b] = r > 0.0f ? r : 0.0f;
}

// ---------------------------------------------------------------------------
// 3a) Zero-pad fc_w (3 x 2048) into fcwPad (16 x 2048) so the WMMA B-matrix
//     loads are unconditional (no EXEC-masked loads in the hot loop).
// ---------------------------------------------------------------------------
__global__ void lcn_pad_fcw(const float* __restrict__ fcw,
                            float* __restrict__ fcwPad) {
  const int i = blockIdx.x * 256 + threadIdx.x;  // 0 .. 16*2048-1
  const int row = i >> 11;                       // / 2048
  fcwPad[i] = (row < 3) ? fcw[i] : 0.0f;
}

// ---------------------------------------------------------------------------
// 3b) Final FC as fp32 WMMA: out(256x3) = act(256x2048) @ fc_w^T + fc_b.
//     One wave per 16-batch tile; D = A(16x4) x B(4x16) + C over K=2048.
//     A layout (32-bit 16x4): lanes 0-15 -> M=0..15, K={0,1} in v0/v1;
//                             lanes 16-31 -> M=0..15, K={2,3}.
//     B layout (4x16): v0 = row K=0 (half0) / K=2 (half1); v1 = K=1 / K=3,
//                      N = lane&15; rows n>=3 read zero-padded fcwPad.
//     4 independent accumulators break the serial D->C dependency chain
//     (f32 WMMA RAW distance ~5); summed element-wise at the end.
//     EXEC is all-ones for every WMMA (divergence only after the loop).
// ---------------------------------------------------------------------------
__global__ void lcn_fc_wmma(const float* __restrict__ act,     // 2048 x 256
                            const float* __restrict__ fcwPad,  // 16 x 2048 (padded)
                            const float* __restrict__ fcb,     // 3
                            float* __restrict__ out) {         // 256 x 3
  const int lane = threadIdx.x;       // 0..31
  const int half = lane >> 4;         // 0/1: K sub-pair
  const int l    = lane & 15;
  const int b0   = blockIdx.x * 16;   // batch tile
  const int b    = b0 + l;            // A: M index = l
  const int n    = l;                 // B/D: N index = l
  const float* __restrict__ fwrow = fcwPad + (size_t)n * 2048;

  v8f c0 = {}, c1 = {}, c2 = {}, c3 = {};
  for (int k0 = 0; k0 < 2048; k0 += 16) {
    #pragma unroll
    for (int j = 0; j < 4; ++j) {
      const int ka = k0 + j * 4 + half * 2;
      v2f a;
      a.x = act[(size_t)(ka + 0) * 256 + b];
      a.y = act[(size_t)(ka + 1) * 256 + b];
      const v2f bm = *(const v2f*)(fwrow + ka);  // unconditional b64 load
      v8f& acc = (j == 0) ? c0 : (j == 1) ? c1 : (j == 2) ? c2 : c3;
      acc = __builtin_amdgcn_wmma_f32_16x16x4_f32(
          /*neg_a=*/false, a, /*neg_b=*/false, bm,
          /*c_mod=*/(short)0, acc, /*reuse_a=*/false, /*reuse_b=*/false);
    }
  }

  // D layout: c[r] holds (M = r + 8*half, N = lane&15); only n < 3 is real.
  if (n < 3) {
    const float bias = fcb[n];
    #pragma unroll
    for (int r = 0; r < 8; ++r) {
      const int m = r + half * 8;
      out[(size_t)(b0 + m) * 3 + n] = (c0[r] + c1[r]) + (c2[r] + c3[r]) + bias;
    }
  }
}

// ---------------------------------------------------------------------------
extern "C" void kernel_launch(void* const* d_in, const int* in_sizes, int n_in,
                              void* d_out, int out_size, void* d_ws, size_t ws_size,
                              hipStream_t stream) {
  // setup_inputs() order: x, fc_w, fc_b, w0,b0,knn0, w1,b1,knn1, w2,b2,knn2
  const float* x    = (const float*)d_in[0];
  const float* fcw  = (const float*)d_in[1];
  const float* fcb  = (const float*)d_in[2];
  const float* w0   = (const float*)d_in[3];
  const float* b0   = (const float*)d_in[4];
  const int*   knn0 = (const int*)  d_in[5];
  const float* w1   = (const float*)d_in[6];
  const float* b1   = (const float*)d_in[7];
  const int*   knn1 = (const int*)  d_in[8];
  const float* w2   = (const float*)d_in[9];
  const float* b2   = (const float*)d_in[10];
  const int*   knn2 = (const int*)  d_in[11];

  float* ws   = (float*)d_ws;
  float* xT   = ws;                          // 16384*256 floats = 16 MB
  float* a1   = xT + (size_t)16384 * 256;    //  8192*256        =  8 MB
  float* a2   = a1 + (size_t)8192 * 256;     //  4096*256        =  4 MB
  float* a3   = a2 + (size_t)4096 * 256;     //  2048*256        =  2 MB
  float* fcwP = a3 + (size_t)2048 * 256;     //    16*2048       = 128 KB

  lcn_transpose<<<dim3(16384 / 32, 256 / 32), dim3(32, 8), 0, stream>>>(x, xT, 16384);
  lcn_layer<<<8192, 256, 0, stream>>>(xT, w0, b0, knn0, a1);
  lcn_layer<<<4096, 256, 0, stream>>>(a1, w1, b1, knn1, a2);
  lcn_layer<<<2048, 256, 0, stream>>>(a2, w2, b2, knn2, a3);
  lcn_pad_fcw<<<(16 * 2048) / 256, 256, 0, stream>>>(fcw, fcwP);
  lcn_fc_wmma<<<16, 32, 0, stream>>>(a3, fcwP, fcb, (float*)d_out);
}